// LBPTransform_62294205661352
// MI455X (gfx1250) — compile-verified
//
#include <hip/hip_runtime.h>
#include <hip/hip_bf16.h>
#include <stdint.h>

// LBP (radius=1, P=8, skimage 'default', zero padding) over NCHW f32.
// Interior tiles: one TDM tensor_load_to_lds DMA per 34x34 halo tile (TENSORcnt).
// Border tiles:   zero-fill + per-lane async global->LDS copies (ASYNCcnt).
// Compute reads the 9-point neighborhood from LDS (stride-1 per lane).

static constexpr int IMG_H = 224;
static constexpr int IMG_W = 224;
static constexpr int TILE  = 32;           // 224 = 7*32, exact tiling
static constexpr int LDIM  = TILE + 2;     // 34 (halo of 1 each side)
static constexpr int LDS_N = LDIM * LDIM;  // 1156 floats = 4624 B

typedef __attribute__((ext_vector_type(4))) unsigned int uint4v;
typedef __attribute__((ext_vector_type(8))) unsigned int uint8v;

__device__ __forceinline__ void lbp_compute_from_lds(
    const float* __restrict__ tile, float* __restrict__ pout,
    int gx0, int gy0, int tid)
{
    const int tx  = tid & (TILE - 1);  // 0..31, stride-1 per lane -> no bank conflicts
    const int ty0 = tid >> 5;          // 0..7

    // skimage offsets for R=1,P=8 (rounded to 8 decimals): A = 0.70710678
    const float A    = 0.70710678f;
    const float Bq   = 0.29289322f;    // 1 - A
    const float w_aa = A * A;
    const float w_ab = A * Bq;
    const float w_bb = Bq * Bq;

    #pragma unroll
    for (int r = 0; r < 4; ++r) {
        const int ry = ty0 + r * 8;  // 0..31 within tile
        const float* rm = &tile[(ry    ) * LDIM + tx];
        const float* rc = &tile[(ry + 1) * LDIM + tx];
        const float* rp = &tile[(ry + 2) * LDIM + tx];

        const float nmm = rm[0], nm0 = rm[1], nmp = rm[2];
        const float n0m = rc[0], c   = rc[1], n0p = rc[2];
        const float npm = rp[0], np0 = rp[1], npp = rp[2];

        const float v1 = w_ab * nm0 + w_aa * nmp + w_bb * c + w_ab * n0p; // up-right
        const float v3 = w_aa * nmm + w_ab * nm0 + w_ab * n0m + w_bb * c; // up-left
        const float v5 = w_ab * n0m + w_bb * c + w_aa * npm + w_ab * np0; // down-left
        const float v7 = w_bb * c + w_ab * n0p + w_ab * np0 + w_aa * npp; // down-right

        int code = 0;
        code |= (n0p >= c) ? 1   : 0;  // p0: ( 0,+1)
        code |= (v1  >= c) ? 2   : 0;  // p1: (-A,+A)
        code |= (nm0 >= c) ? 4   : 0;  // p2: (-1, 0)
        code |= (v3  >= c) ? 8   : 0;  // p3: (-A,-A)
        code |= (n0m >= c) ? 16  : 0;  // p4: ( 0,-1)
        code |= (v5  >= c) ? 32  : 0;  // p5: (+A,-A)
        code |= (np0 >= c) ? 64  : 0;  // p6: (+1, 0)
        code |= (v7  >= c) ? 128 : 0;  // p7: (+A,+A)

        __builtin_nontemporal_store((float)code,
                                    &pout[(gy0 + ry) * IMG_W + gx0 + tx]);
    }
}

// ---------------------------------------------------------------------------
// Interior tiles (5x5 of the 7x7 grid): whole halo in-bounds -> one TDM DMA.
// ---------------------------------------------------------------------------
__global__ __launch_bounds__(256)
void lbp_interior_kernel(const float* __restrict__ x, float* __restrict__ out) {
    __shared__ float tile[LDS_N];

    const int bid   = blockIdx.x;
    const int plane = bid / 25;
    const int t     = bid % 25;
    const int gy0   = (t / 5 + 1) * TILE;   // tile rows 1..5
    const int gx0   = (t % 5 + 1) * TILE;   // tile cols 1..5

    const float* __restrict__ px   = x   + (size_t)plane * (IMG_H * IMG_W);
    float*       __restrict__ pout = out + (size_t)plane * (IMG_H * IMG_W);

    if (threadIdx.x < 32) {  // single wave issues the DMA (TDM ignores EXEC)
        const float* gp = px + ((gy0 - 1) * IMG_W + (gx0 - 1));
        const unsigned long long ga = (unsigned long long)(size_t)gp;
        const unsigned lds0 = (unsigned)(size_t)(&tile[0]);

        // D# group 0: count=1 | lds_addr | global_addr[56:0] | type=2
        uint4v g0;
        g0.x = 1u;
        g0.y = lds0;
        g0.z = (unsigned)(ga & 0xFFFFFFFFull);
        g0.w = (unsigned)((ga >> 32) & 0x1FFFFFFull) | (2u << 30);

        // D# group 1: data_size=2 (4B); tensor dims huge (no OOB on interior);
        // tile 34x34; tensor_dim0_stride = 224 elements; no pad/iterate/multicast.
        constexpr unsigned TD0 = 1u << 24;
        constexpr unsigned TD1 = 1u << 24;
        uint8v g1;
        g1[0] = 2u << 16;                                        // data_size
        g1[1] = (TD0 & 0xFFFFu) << 16;                           // tensor_dim0 lo
        g1[2] = ((TD0 >> 16) & 0xFFFFu) | ((TD1 & 0xFFFFu) << 16);
        g1[3] = ((TD1 >> 16) & 0xFFFFu) | ((unsigned)LDIM << 16);// tile_dim0=34
        g1[4] = (unsigned)LDIM;                                  // tile_dim1=34, tile_dim2=0
        g1[5] = (unsigned)IMG_W;                                 // dim0_stride lo = 224
        g1[6] = 0u;                                              // dim0_stride hi | dim1_stride lo
        g1[7] = 0u;                                              // dim1_stride hi

        asm volatile("tensor_load_to_lds %0, %1"
                     :
                     : "s"(g0), "s"(g1)
                     : "memory");
        __builtin_amdgcn_s_wait_tensorcnt(0);
    }
    __syncthreads();

    lbp_compute_from_lds(tile, pout, gx0, gy0, (int)threadIdx.x);
}

// ---------------------------------------------------------------------------
// Border tiles (ring of 24): zero-fill LDS (constant-0 image border), then
// predicated per-lane async global->LDS copies for the in-bounds region.
// ---------------------------------------------------------------------------
__global__ __launch_bounds__(256)
void lbp_border_kernel(const float* __restrict__ x, float* __restrict__ out) {
    __shared__ float tile[LDS_N];

    const int bid   = blockIdx.x;
    const int plane = bid / 24;
    const int b     = bid % 24;

    int tyT, txT;  // ring enumeration of the 7x7 tile grid border
    if (b < 7)       { tyT = 0;          txT = b;     }
    else if (b < 14) { tyT = 6;          txT = b - 7; }
    else if (b < 19) { tyT = b - 14 + 1; txT = 0;     }
    else             { tyT = b - 19 + 1; txT = 6;     }
    const int gy0 = tyT * TILE;
    const int gx0 = txT * TILE;

    const float* __restrict__ px   = x   + (size_t)plane * (IMG_H * IMG_W);
    float*       __restrict__ pout = out + (size_t)plane * (IMG_H * IMG_W);

    const int tid = threadIdx.x;

    // 1) zero-fill: implements constant-0 padding outside the image
    #pragma unroll
    for (int k = tid; k < LDS_N; k += 256) tile[k] = 0.0f;
    __syncthreads();

    // 2) async copy valid region global -> LDS (ASYNCcnt path)
    for (int k = tid; k < LDS_N; k += 256) {
        const int li = k / LDIM;
        const int lj = k - li * LDIM;
        const int gy = gy0 - 1 + li;
        const int gx = gx0 - 1 + lj;
        if ((unsigned)gy < (unsigned)IMG_H && (unsigned)gx < (unsigned)IMG_W) {
            const unsigned lds_addr = (unsigned)(size_t)(&tile[k]);
            const unsigned goff     = (unsigned)((gy * IMG_W + gx) * 4);
            asm volatile("global_load_async_to_lds_b32 %0, %1, %2"
                         :
                         : "v"(lds_addr), "v"(goff), "s"(px)
                         : "memory");
        }
    }
    asm volatile("s_wait_asynccnt 0" ::: "memory");
    __syncthreads();

    lbp_compute_from_lds(tile, pout, gx0, gy0, tid);
}

extern "C" void kernel_launch(void* const* d_in, const int* in_sizes, int n_in,
                              void* d_out, int out_size, void* d_ws, size_t ws_size,
                              hipStream_t stream) {
    (void)n_in; (void)d_ws; (void)ws_size; (void)out_size;
    const float* x = (const float*)d_in[0];
    float* out = (float*)d_out;

    const int planes = in_sizes[0] / (IMG_H * IMG_W);  // 16*64 = 1024

    dim3 gridI((unsigned)(planes * 25));  // interior tiles via TDM
    dim3 gridB((unsigned)(planes * 24));  // border ring via async copies
    lbp_interior_kernel<<<gridI, 256, 0, stream>>>(x, out);
    lbp_border_kernel  <<<gridB, 256, 0, stream>>>(x, out);
}